// MyTransConvLayer_70806830842559
// MI455X (gfx1250) — compile-verified
//
#include <hip/hip_runtime.h>
#include <math.h>

// Problem constants (from reference): N=32768, C=1024, H=8, M=64, O=H*M=512, NB=4
#define NN_TOK 32768
#define CC     1024
#define OO     512
#define HH     8
#define MM     64
#define KVS_SEG 32   // split-N segments for kvs partial reduction

typedef float v2f __attribute__((ext_vector_type(2)));
typedef float v8f __attribute__((ext_vector_type(8)));

__device__ __forceinline__ v8f wmma_f32(v2f a, v2f b, v8f c) {
    // V_WMMA_F32_16X16X4_F32 : D = A(16x4 f32) * B(4x16 f32) + C(16x16 f32)
    return __builtin_amdgcn_wmma_f32_16x16x4_f32(
        /*neg_a=*/false, a, /*neg_b=*/false, b,
        /*c_mod=*/(short)0, c, /*reuse_a=*/false, /*reuse_b=*/false);
}

// ---------------------------------------------------------------------------
// Kernel 1: HO linear.  out[n,o] = poly(ow; ips[0..3])  where
// ips[i] = x[n, 256i:256(i+1)] . W[o, 256i:256(i+1)]
// grid = (N/64, O/64), block = 128 (4 waves).
// Wave w owns n-rows [n0+16w, n0+16w+16) and a 16x64 output strip:
// 4 o-subtiles x 4 K-segment partial accumulators = 16 v8f accumulators.
// The A fragment is loaded once per k-step and reused for all 4 o-subtiles.
// ---------------------------------------------------------------------------
__global__ void ho_linear_kernel(const float* __restrict__ X,
                                 const float* __restrict__ W,
                                 const float* __restrict__ ow,
                                 float* __restrict__ out)
{
    const int wave  = threadIdx.x >> 5;
    const int lane  = threadIdx.x & 31;
    const int half  = lane >> 4;      // 0: K=0,1 ; 1: K=2,3
    const int i16   = lane & 15;
    const int n0    = blockIdx.x * 64 + wave * 16;
    const int o0    = blockIdx.y * 64;

    const float* arow = X + (size_t)(n0 + i16) * CC;   // A row m = i16
    const float* brow0 = W + (size_t)(o0 +  0 + i16) * CC;  // B col o = i16 (W^T)
    const float* brow1 = W + (size_t)(o0 + 16 + i16) * CC;
    const float* brow2 = W + (size_t)(o0 + 32 + i16) * CC;
    const float* brow3 = W + (size_t)(o0 + 48 + i16) * CC;

    v8f acc[4][4] = {};   // [K-segment][o-subtile]
#pragma unroll
    for (int ib = 0; ib < 4; ++ib) {
        const int kbeg = ib * 256;
        // prefetch next K-segment of the streaming x row (speculative; OOB dropped)
        __builtin_prefetch(arow + kbeg + 256, 0, 1);
#pragma unroll 8
        for (int k = kbeg; k < kbeg + 256; k += 4) {
            v2f a  = *(const v2f*)(arow  + k + 2 * half);  // (m, k+2h), (m, k+2h+1)
            v2f b0 = *(const v2f*)(brow0 + k + 2 * half);  // (k+2h, o), (k+2h+1, o)
            v2f b1 = *(const v2f*)(brow1 + k + 2 * half);
            v2f b2 = *(const v2f*)(brow2 + k + 2 * half);
            v2f b3 = *(const v2f*)(brow3 + k + 2 * half);
            acc[ib][0] = wmma_f32(a, b0, acc[ib][0]);
            acc[ib][1] = wmma_f32(a, b1, acc[ib][1]);
            acc[ib][2] = wmma_f32(a, b2, acc[ib][2]);
            acc[ib][3] = wmma_f32(a, b3, acc[ib][3]);
        }
    }

    const float w0 = ow[0], w1 = ow[1], w2 = ow[2], w3 = ow[3], w4 = ow[4];
#pragma unroll
    for (int ot = 0; ot < 4; ++ot) {
#pragma unroll
        for (int r = 0; r < 8; ++r) {
            float a = acc[0][ot][r], b = acc[1][ot][r];
            float c = acc[2][ot][r], d = acc[3][ot][r];
            float ab = a * b, cd = c * d;
            float e1 = (a + b) + (c + d);
            float e2 = ab + cd + a * c + a * d + b * c + b * d;
            float e3 = ab * (c + d) + cd * (a + b);
            float e4 = ab * cd;
            float res = w0 + w1 * e1 + w2 * e2 + w3 * e3 + w4 * e4;
            out[(size_t)(n0 + r + 8 * half) * OO + o0 + ot * 16 + i16] = res;
        }
    }
}

// ---------------------------------------------------------------------------
// Kernel 2a: partial sum of squares (deterministic two-pass)
// ---------------------------------------------------------------------------
__global__ void sumsq_kernel(const float* __restrict__ x, int n,
                             float* __restrict__ partial)
{
    __shared__ float sm[256];
    float s = 0.f;
    for (int i = blockIdx.x * 256 + threadIdx.x; i < n; i += 256 * gridDim.x) {
        float v = x[i];
        s += v * v;
    }
    sm[threadIdx.x] = s;
    __syncthreads();
    for (int off = 128; off > 0; off >>= 1) {
        if (threadIdx.x < off) sm[threadIdx.x] += sm[threadIdx.x + off];
        __syncthreads();
    }
    if (threadIdx.x == 0) partial[blockIdx.x] = sm[0];
}

// Kernel 2b: finalize s = 1/(||qs|| * ||ks||)
__global__ void scale_kernel(const float* __restrict__ pq,
                             const float* __restrict__ pk,
                             float* __restrict__ s_out)
{
    __shared__ float sq[256], sk[256];
    sq[threadIdx.x] = pq[threadIdx.x];
    sk[threadIdx.x] = pk[threadIdx.x];
    __syncthreads();
    for (int off = 128; off > 0; off >>= 1) {
        if (threadIdx.x < off) {
            sq[threadIdx.x] += sq[threadIdx.x + off];
            sk[threadIdx.x] += sk[threadIdx.x + off];
        }
        __syncthreads();
    }
    if (threadIdx.x == 0) s_out[0] = rsqrtf(sq[0]) * rsqrtf(sk[0]);
}

// ---------------------------------------------------------------------------
// Kernel 3: kvs partials.  kvs[h,m,d] = sum_n ks[n,h*64+m]*vs[n,h*64+d]
// grid = (H, 4 m-tiles, KVS_SEG), block = 128 (wave = d-tile). Split-N.
// ---------------------------------------------------------------------------
__global__ void kvs_partial_kernel(const float* __restrict__ ks,
                                   const float* __restrict__ vs,
                                   float* __restrict__ kvsp)
{
    const int wave = threadIdx.x >> 5;      // d-tile
    const int lane = threadIdx.x & 31;
    const int half = lane >> 4;
    const int i16  = lane & 15;
    const int h    = blockIdx.x;
    const int mt   = blockIdx.y;
    const int seg  = blockIdx.z;

    const int mcol = h * 64 + mt * 16 + i16;    // A column in ks (k = n)
    const int dcol = h * 64 + wave * 16 + i16;  // B column in vs

    v8f acc = {};
    const int nbeg = seg * (NN_TOK / KVS_SEG);
#pragma unroll 4
    for (int nn = nbeg; nn < nbeg + NN_TOK / KVS_SEG; nn += 4) {
        v2f a, b;
        a.x = ks[(size_t)(nn + 2 * half)     * OO + mcol];
        a.y = ks[(size_t)(nn + 2 * half + 1) * OO + mcol];
        b.x = vs[(size_t)(nn + 2 * half)     * OO + dcol];
        b.y = vs[(size_t)(nn + 2 * half + 1) * OO + dcol];
        acc = wmma_f32(a, b, acc);
    }

    float* dst = kvsp + ((size_t)seg * HH + h) * (MM * MM);
#pragma unroll
    for (int r = 0; r < 8; ++r)
        dst[(mt * 16 + r + 8 * half) * MM + wave * 16 + i16] = acc[r];
}

// Kernel 4: reduce kvs partials over segments (fixed order)
__global__ void kvs_reduce_kernel(const float* __restrict__ kvsp,
                                  float* __restrict__ kvs)
{
    int i = blockIdx.x * blockDim.x + threadIdx.x;  // < H*64*64
    float s = 0.f;
    for (int seg = 0; seg < KVS_SEG; ++seg)
        s += kvsp[(size_t)seg * (HH * MM * MM) + i];
    kvs[i] = s;
}

// Kernel 5: ks_sum[o] = sum_n ks[n,o]   (one block per column, fixed order)
__global__ void colsum_kernel(const float* __restrict__ ks,
                              float* __restrict__ kssum)
{
    __shared__ float sm[256];
    const int col = blockIdx.x;
    float s = 0.f;
    for (int n = threadIdx.x; n < NN_TOK; n += 256)
        s += ks[(size_t)n * OO + col];
    sm[threadIdx.x] = s;
    __syncthreads();
    for (int off = 128; off > 0; off >>= 1) {
        if (threadIdx.x < off) sm[threadIdx.x] += sm[threadIdx.x + off];
        __syncthreads();
    }
    if (threadIdx.x == 0) kssum[col] = sm[0];
}

// ---------------------------------------------------------------------------
// Kernel 6: normalizer[n,h] = poly2(softmax(ow_dot); s*ip[0..3]) + N
// where ip[i] = qs[n, h*64+16i : +16] . ks_sum[h*64+16i : +16]
// ---------------------------------------------------------------------------
__global__ void normalizer_kernel(const float* __restrict__ qs,
                                  const float* __restrict__ kssum,
                                  const float* __restrict__ owdot,
                                  const float* __restrict__ sptr,
                                  float* __restrict__ nrm)
{
    int idx = blockIdx.x * blockDim.x + threadIdx.x;  // < N*H
    int h = idx & 7, n = idx >> 3;
    float o0 = owdot[0], o1 = owdot[1], o2 = owdot[2];
    float mx = fmaxf(o0, fmaxf(o1, o2));
    float e0 = expf(o0 - mx), e1 = expf(o1 - mx), e2 = expf(o2 - mx);
    float inv = 1.f / (e0 + e1 + e2);
    float w0 = e0 * inv, w1 = e1 * inv, w2 = e2 * inv;

    const float* q = qs + (size_t)n * OO + h * 64;
    const float* k = kssum + h * 64;
    float s = sptr[0];
    float ip[4];
#pragma unroll
    for (int i = 0; i < 4; ++i) {
        float t = 0.f;
#pragma unroll
        for (int j = 0; j < 16; ++j) t += q[i * 16 + j] * k[i * 16 + j];
        ip[i] = t * s;
    }
    float se = ip[0] + ip[1] + ip[2] + ip[3];
    float p2 = ip[0] * ip[1] + ip[0] * ip[2] + ip[0] * ip[3] +
               ip[1] * ip[2] + ip[1] * ip[3] + ip[2] * ip[3];
    nrm[idx] = w0 + w1 * se + w2 * p2 + (float)NN_TOK;
}

// ---------------------------------------------------------------------------
// Kernel 7: final.  For each (n-tile, d-tile) wave, loop heads:
//   ip[i][n,d] = sum_{m in blk i} qs[n,h*64+m]*kvs[h,m,d]   (WMMA, 4 accs)
//   num = poly2(w; s*ip) + N*vs[n,h*64+d]
//   out[n,d] = (1/H) * sum_h num / nrm[n,h]
// grid = N/16, block = 128 (4 waves = 4 d-tiles of 16).
// ---------------------------------------------------------------------------
__global__ void attn_final_kernel(const float* __restrict__ qs,
                                  const float* __restrict__ vs,
                                  const float* __restrict__ kvs,
                                  const float* __restrict__ nrm,
                                  const float* __restrict__ owdot,
                                  const float* __restrict__ sptr,
                                  float* __restrict__ out)
{
    const int wave = threadIdx.x >> 5;  // d-tile
    const int lane = threadIdx.x & 31;
    const int half = lane >> 4;
    const int i16  = lane & 15;
    const int n0   = blockIdx.x * 16;
    const int d    = wave * 16 + i16;

    float o0 = owdot[0], o1 = owdot[1], o2 = owdot[2];
    float mx = fmaxf(o0, fmaxf(o1, o2));
    float e0 = expf(o0 - mx), e1 = expf(o1 - mx), e2 = expf(o2 - mx);
    float inv = 1.f / (e0 + e1 + e2);
    float w0 = e0 * inv, w1 = e1 * inv, w2 = e2 * inv;
    float s = sptr[0];

    float res[8] = {0.f, 0.f, 0.f, 0.f, 0.f, 0.f, 0.f, 0.f};

    for (int h = 0; h < HH; ++h) {
        const float* arow = qs + (size_t)(n0 + i16) * OO + h * 64;
        const float* kb   = kvs + (size_t)h * (MM * MM);
        v8f acc[4] = {};
#pragma unroll
        for (int i = 0; i < 4; ++i) {
#pragma unroll
            for (int k = i * 16; k < i * 16 + 16; k += 4) {
                v2f a = *(const v2f*)(arow + k + 2 * half);
                v2f b;
                b.x = kb[(k + 2 * half)     * MM + d];
                b.y = kb[(k + 2 * half + 1) * MM + d];
                acc[i] = wmma_f32(a, b, acc[i]);
            }
        }
#pragma unroll
        for (int r = 0; r < 8; ++r) {
            const int n = n0 + r + 8 * half;
            float i0 = acc[0][r] * s, i1 = acc[1][r] * s;
            float i2 = acc[2][r] * s, i3 = acc[3][r] * s;
            float se = i0 + i1 + i2 + i3;
            float p2 = i0 * i1 + i0 * i2 + i0 * i3 + i1 * i2 + i1 * i3 + i2 * i3;
            float num = w0 + w1 * se + w2 * p2
                      + (float)NN_TOK * vs[(size_t)n * OO + h * 64 + d];
            res[r] += num / nrm[n * HH + h];
        }
    }
#pragma unroll
    for (int r = 0; r < 8; ++r)
        out[(size_t)(n0 + r + 8 * half) * MM + d] = res[r] * (1.0f / HH);
}

// ---------------------------------------------------------------------------
extern "C" void kernel_launch(void* const* d_in, const int* in_sizes, int n_in,
                              void* d_out, int out_size, void* d_ws, size_t ws_size,
                              hipStream_t stream) {
    const float* query  = (const float*)d_in[0];
    const float* source = (const float*)d_in[1];
    const float* Wq     = (const float*)d_in[2];
    const float* Wk     = (const float*)d_in[3];
    const float* Wv     = (const float*)d_in[4];
    const float* ow_q   = (const float*)d_in[5];
    const float* ow_k   = (const float*)d_in[6];
    const float* ow_v   = (const float*)d_in[7];
    const float* ow_dot = (const float*)d_in[8];
    float* out = (float*)d_out;

    float* w = (float*)d_ws;
    const size_t NO = (size_t)NN_TOK * OO;
    float* qs    = w;
    float* ks    = w + NO;
    float* vsb   = w + 2 * NO;
    float* pq    = w + 3 * NO;            // 256
    float* pk    = pq + 256;              // 256
    float* sptr  = pk + 256;              // 1 (padded to 256)
    float* kvsp  = sptr + 256;            // KVS_SEG * H * 64 * 64
    float* kvsF  = kvsp + (size_t)KVS_SEG * HH * MM * MM;  // H*64*64
    float* kssum = kvsF + (size_t)HH * MM * MM;            // 512
    float* nrm   = kssum + OO;            // N*H

    dim3 gLin(NN_TOK / 64, OO / 64);
    ho_linear_kernel<<<gLin, 128, 0, stream>>>(query,  Wq, ow_q, qs);
    ho_linear_kernel<<<gLin, 128, 0, stream>>>(source, Wk, ow_k, ks);
    ho_linear_kernel<<<gLin, 128, 0, stream>>>(source, Wv, ow_v, vsb);

    sumsq_kernel<<<256, 256, 0, stream>>>(qs, (int)NO, pq);
    sumsq_kernel<<<256, 256, 0, stream>>>(ks, (int)NO, pk);
    scale_kernel<<<1, 256, 0, stream>>>(pq, pk, sptr);

    kvs_partial_kernel<<<dim3(HH, 4, KVS_SEG), 128, 0, stream>>>(ks, vsb, kvsp);
    kvs_reduce_kernel<<<(HH * MM * MM) / 256, 256, 0, stream>>>(kvsp, kvsF);
    colsum_kernel<<<OO, 256, 0, stream>>>(ks, kssum);

    normalizer_kernel<<<(NN_TOK * HH) / 256, 256, 0, stream>>>(qs, kssum, ow_dot, sptr, nrm);

    attn_final_kernel<<<NN_TOK / 16, 128, 0, stream>>>(qs, vsb, kvsF, nrm, ow_dot, sptr, out);
}